// HNN1DWaveSeparable_43533788512669
// MI455X (gfx1250) — compile-verified
//
#include <hip/hip_runtime.h>

typedef __attribute__((ext_vector_type(2))) float v2f;
typedef __attribute__((ext_vector_type(8))) float v8f;

// ---------------------------------------------------------------------------
// K1: h1/s1 = tanh(z@W1 + b1), z = [x,q,p].  Coalesced column-walk over W1.
// ---------------------------------------------------------------------------
__global__ void k_layer1(const float* __restrict__ x, const float* __restrict__ q,
                         const float* __restrict__ p, const float* __restrict__ W1,
                         const float* __restrict__ b1, float* __restrict__ h1,
                         float* __restrict__ s1, int n, int hid) {
    int j = blockIdx.x * blockDim.x + threadIdx.x;
    if (j >= hid) return;
    float a = b1[j];
    const float* w = W1 + j;
    for (int k = 0; k < n; ++k) a += x[k] * w[(size_t)k * hid];
    w += (size_t)n * hid;
    for (int k = 0; k < n; ++k) a += q[k] * w[(size_t)k * hid];
    w += (size_t)n * hid;
    for (int k = 0; k < n; ++k) a += p[k] * w[(size_t)k * hid];
    float h = tanhf(a);
    h1[j] = h;
    s1[j] = 1.0f - h * h;
}

// ---------------------------------------------------------------------------
// K2: a2 = h1@W2 + b2; u2 = w3*s2; e = -2*w3*h2*s2
// ---------------------------------------------------------------------------
__global__ void k_layer2(const float* __restrict__ h1, const float* __restrict__ W2,
                         const float* __restrict__ b2, const float* __restrict__ w3,
                         float* __restrict__ u2, float* __restrict__ ev, int hid) {
    int j = blockIdx.x * blockDim.x + threadIdx.x;
    if (j >= hid) return;
    float a = b2[j];
    const float* w = W2 + j;
    for (int k = 0; k < hid; ++k) a += h1[k] * w[(size_t)k * hid];
    float h = tanhf(a);
    float s = 1.0f - h * h;
    u2[j] = w3[j] * s;
    ev[j] = -2.0f * w3[j] * h * s;
}

// ---------------------------------------------------------------------------
// K3: v_j = W2[j,:]·u2 ; d_j = -2*v_j*h1_j*s1_j.  One block per row.
// ---------------------------------------------------------------------------
__global__ void k_vd(const float* __restrict__ W2, const float* __restrict__ u2,
                     const float* __restrict__ h1, const float* __restrict__ s1,
                     float* __restrict__ dvec, int hid) {
    int j = blockIdx.x;
    __shared__ float red[256];
    float s = 0.0f;
    const float* row = W2 + (size_t)j * hid;
    for (int k = threadIdx.x; k < hid; k += blockDim.x) s += row[k] * u2[k];
    red[threadIdx.x] = s;
    __syncthreads();
    for (int w = 128; w > 0; w >>= 1) {
        if ((int)threadIdx.x < w) red[threadIdx.x] += red[threadIdx.x + w];
        __syncthreads();
    }
    if (threadIdx.x == 0) dvec[j] = -2.0f * red[0] * h1[j] * s1[j];
}

// ---------------------------------------------------------------------------
// K4: W2T = W2^T (so both GEMMs share one row-major-B WMMA kernel)
// ---------------------------------------------------------------------------
__global__ void k_transpose(const float* __restrict__ in, float* __restrict__ out, int hid) {
    __shared__ float tile[32][33];
    int bx = blockIdx.x * 32, by = blockIdx.y * 32;
    int tx = threadIdx.x, ty = threadIdx.y;  // 32 x 8
    for (int r = 0; r < 32; r += 8)
        tile[ty + r][tx] = in[(size_t)(by + ty + r) * hid + bx + tx];
    __syncthreads();
    for (int r = 0; r < 32; r += 8)
        out[(size_t)(bx + ty + r) * hid + by + tx] = tile[tx][ty + r];
}

// ---------------------------------------------------------------------------
// K5: dst[r,c] = src[r,c] * scale[c]   (vectorized float4; hid % 4 == 0)
// Hoists the per-K scaling out of the GEMM so the GEMM can stage tiles with
// raw async DMA copies.
// ---------------------------------------------------------------------------
__global__ void k_scale_cols(const float* __restrict__ src, const float* __restrict__ scale,
                             float* __restrict__ dst, int rows, int hid) {
    size_t idx = ((size_t)blockIdx.x * blockDim.x + threadIdx.x) * 4;
    size_t total = (size_t)rows * hid;
    if (idx >= total) return;
    int col = (int)(idx % hid);
    float4 v = *(const float4*)(src + idx);
    v.x *= scale[col];
    v.y *= scale[col + 1];
    v.z *= scale[col + 2];
    v.w *= scale[col + 3];
    *(float4*)(dst + idx) = v;
}

// ---------------------------------------------------------------------------
// K6: WMMA GEMM  C = A @ B   (M=N=K multiples of 128/16)
// 128x128 block tile, BK=16, double-buffered LDS, 8 wave32 waves, each wave
// computes 32x64 via a 2x4 grid of V_WMMA_F32_16X16X4_F32.
// Staging uses GLOBAL_LOAD_ASYNC_TO_LDS_B128 (ASYNCcnt): 4 instructions per
// wave per stage, no VGPR staging, next stage's DMA overlaps current WMMAs.
// LDS pads (A rows -> 20 words, B rows -> 136 words) keep both the 16B async
// chunks aligned and the f32 16x16x4 fragment reads bank-conflict free.
// ---------------------------------------------------------------------------
#define GBM 128
#define GBN 128
#define GBK 16
#define PADA 20
#define PADB 136

__global__ __launch_bounds__(256) void k_gemm_async(
    const float* __restrict__ A, const float* __restrict__ B,
    float* __restrict__ C, int K, int lda, int ldb, int ldc) {
    __shared__ float As[2][GBM * PADA];
    __shared__ float Bs[2][GBK * PADB];

    const int tid  = threadIdx.x;
    const int wid  = tid >> 5;      // wave32
    const int lane = tid & 31;
    const int wm   = wid & 3;       // 0..3 -> 32-row band
    const int wn   = wid >> 2;      // 0..1 -> 64-col band
    const int l16  = lane & 15;
    const int lh   = lane >> 4;     // 0/1 -> K-half of fragment

    const int i0 = blockIdx.y * GBM;
    const int n0 = blockIdx.x * GBN;

    // LDS byte offsets of the staging buffers (generic LDS pointers carry the
    // LDS offset in bits [31:0] on gfx12xx).
    const unsigned asOff[2] = {(unsigned)(size_t)&As[0][0], (unsigned)(size_t)&As[1][0]};
    const unsigned bsOff[2] = {(unsigned)(size_t)&Bs[0][0], (unsigned)(size_t)&Bs[1][0]};

    // Async staging: 512 chunks of 16B per tile (A and B each); each wave
    // issues 2 A-chunks + 2 B-chunks per stage.
    auto issue_stage = [&](int jblk, int buf) {
#pragma unroll
        for (int t = 0; t < 2; ++t) {
            const int q = wid * 64 + t * 32 + lane;  // 0..511
            // A chunk: row r (0..127), 4 floats starting at col ca
            const int r  = q >> 2;
            const int ca = (q & 3) << 2;
            const float* ga = A + (size_t)(i0 + r) * lda + jblk + ca;
            const unsigned la = asOff[buf] + (unsigned)((r * PADA + ca) * 4);
            asm volatile("global_load_async_to_lds_b128 %0, %1, off"
                         :: "v"(la), "v"(ga) : "memory");
            // B chunk: row kk (0..15), 4 floats starting at col cb
            const int kk = q >> 5;
            const int cb = (q & 31) << 2;
            const float* gb = B + (size_t)(jblk + kk) * ldb + n0 + cb;
            const unsigned lb = bsOff[buf] + (unsigned)((kk * PADB + cb) * 4);
            asm volatile("global_load_async_to_lds_b128 %0, %1, off"
                         :: "v"(lb), "v"(gb) : "memory");
        }
    };

    v8f acc[2][4];
#pragma unroll
    for (int mi = 0; mi < 2; ++mi)
#pragma unroll
        for (int ni = 0; ni < 4; ++ni)
            acc[mi][ni] = (v8f){0.f, 0.f, 0.f, 0.f, 0.f, 0.f, 0.f, 0.f};

    auto compute = [&](int buf) {
        const float* as = As[buf];
        const float* bs = Bs[buf];
#pragma unroll
        for (int ks = 0; ks < 4; ++ks) {
            const int kk = ks * 4 + 2 * lh;  // this lane-half's K pair
            v2f af[2];
            v2f bf[4];
#pragma unroll
            for (int mi = 0; mi < 2; ++mi) {
                const int row = wm * 32 + mi * 16 + l16;
                af[mi].x = as[row * PADA + kk];
                af[mi].y = as[row * PADA + kk + 1];
            }
#pragma unroll
            for (int ni = 0; ni < 4; ++ni) {
                const int col = wn * 64 + ni * 16 + l16;
                bf[ni].x = bs[kk * PADB + col];
                bf[ni].y = bs[(kk + 1) * PADB + col];
            }
#pragma unroll
            for (int mi = 0; mi < 2; ++mi)
#pragma unroll
                for (int ni = 0; ni < 4; ++ni)
                    acc[mi][ni] = __builtin_amdgcn_wmma_f32_16x16x4_f32(
                        false, af[mi], false, bf[ni], (short)0, acc[mi][ni],
                        false, false);
        }
    };

    const int nstages = K / GBK;
    issue_stage(0, 0);
    for (int s = 0; s < nstages; ++s) {
        // Wait for this wave's stage-s DMA, then barrier so all waves' tiles
        // are visible (and all readers of the other buffer are done).
        asm volatile("s_wait_asynccnt 0x0" ::: "memory");
        __syncthreads();
        if (s + 1 < nstages) issue_stage((s + 1) * GBK, (s + 1) & 1);
        compute(s & 1);  // WMMAs overlap the in-flight stage-(s+1) DMA
    }

    // epilogue: f32 16x16 D layout — VGPR v: M = v + 8*(lane/16), N = lane%16
#pragma unroll
    for (int mi = 0; mi < 2; ++mi)
#pragma unroll
        for (int ni = 0; ni < 4; ++ni) {
            const int row0 = i0 + wm * 32 + mi * 16 + lh * 8;
            const int col  = n0 + wn * 64 + ni * 16 + l16;
#pragma unroll
            for (int v = 0; v < 8; ++v)
                C[(size_t)(row0 + v) * ldc + col] = acc[mi][ni][v];
        }
}

// ---------------------------------------------------------------------------
// K7: out[i]   =  sum_j W1p[i,j] * (s1_j*DV[i,j] + d_j*W1x[i,j])
//     out[N+i] = -sum_j W1q[i,j] * (s1_j*DV[i,j] + d_j*W1x[i,j])
// ---------------------------------------------------------------------------
__global__ void k_finalize(const float* __restrict__ W1, const float* __restrict__ DV,
                           const float* __restrict__ s1, const float* __restrict__ dvec,
                           float* __restrict__ out, int n, int hid) {
    int i = blockIdx.x;
    __shared__ float redp[256];
    __shared__ float redq[256];
    const float* Ai  = W1 + (size_t)i * hid;            // x-block row
    const float* Bi  = W1 + (size_t)(n + i) * hid;      // q-block row
    const float* Ci  = W1 + (size_t)(2 * n + i) * hid;  // p-block row
    const float* dvi = DV + (size_t)i * hid;
    float sp = 0.f, sq = 0.f;
    for (int j = threadIdx.x; j < hid; j += blockDim.x) {
        float t = s1[j] * dvi[j] + dvec[j] * Ai[j];
        sp += Ci[j] * t;
        sq += Bi[j] * t;
    }
    redp[threadIdx.x] = sp;
    redq[threadIdx.x] = sq;
    __syncthreads();
    for (int w = 128; w > 0; w >>= 1) {
        if ((int)threadIdx.x < w) {
            redp[threadIdx.x] += redp[threadIdx.x + w];
            redq[threadIdx.x] += redq[threadIdx.x + w];
        }
        __syncthreads();
    }
    if (threadIdx.x == 0) {
        out[i]     = redp[0];   // dH_dp_dx
        out[n + i] = -redq[0];  // -dH_dq_dx
    }
}

// ---------------------------------------------------------------------------
extern "C" void kernel_launch(void* const* d_in, const int* in_sizes, int n_in,
                              void* d_out, int out_size, void* d_ws, size_t ws_size,
                              hipStream_t stream) {
    const float* x  = (const float*)d_in[0];
    const float* q  = (const float*)d_in[1];
    const float* p  = (const float*)d_in[2];
    const float* W1 = (const float*)d_in[3];
    const float* b1 = (const float*)d_in[4];
    const float* W2 = (const float*)d_in[5];
    const float* b2 = (const float*)d_in[6];
    const float* w3 = (const float*)d_in[7];
    // b3 (d_in[8]) drops out of all derivatives.
    float* out = (float*)d_out;

    const int n   = in_sizes[0];   // 1024
    const int hid = in_sizes[4];   // 1024

    float* ws   = (float*)d_ws;
    float* h1   = ws;
    float* s1   = ws + hid;
    float* u2   = ws + 2 * hid;
    float* ev   = ws + 3 * hid;
    float* dvec = ws + 4 * hid;
    float* W2T  = ws + 8192;
    float* DA2  = W2T + (size_t)hid * hid;
    float* DV   = DA2 + (size_t)n * hid;
    float* Ahat = DV + (size_t)n * hid;

    k_layer1<<<(hid + 255) / 256, 256, 0, stream>>>(x, q, p, W1, b1, h1, s1, n, hid);
    k_layer2<<<(hid + 255) / 256, 256, 0, stream>>>(h1, W2, b2, w3, u2, ev, hid);
    k_vd<<<hid, 256, 0, stream>>>(W2, u2, h1, s1, dvec, hid);
    k_transpose<<<dim3(hid / 32, hid / 32), dim3(32, 8), 0, stream>>>(W2, W2T, hid);

    const int nquads = (n * hid) / 4;
    dim3 gg(hid / GBN, n / GBM);

    // Ahat = W1x . s1 ; DA2 = Ahat @ W2
    k_scale_cols<<<(nquads + 255) / 256, 256, 0, stream>>>(W1, s1, Ahat, n, hid);
    k_gemm_async<<<gg, 256, 0, stream>>>(Ahat, W2, DA2, hid, hid, hid, hid);

    // DA2 .= e ; DV = DA2 @ W2^T
    k_scale_cols<<<(nquads + 255) / 256, 256, 0, stream>>>(DA2, ev, DA2, n, hid);
    k_gemm_async<<<gg, 256, 0, stream>>>(DA2, W2T, DV, hid, hid, hid, hid);

    k_finalize<<<n, 256, 0, stream>>>(W1, DV, s1, dvec, out, n, hid);
}